// TAHIN_52458730553647
// MI455X (gfx1250) — compile-verified
//
#include <hip/hip_runtime.h>
#include <hip/hip_bf16.h>

typedef __attribute__((ext_vector_type(2))) float v2f;
typedef __attribute__((ext_vector_type(8))) float v8f;

static constexpr int FD = 64;   // feature dim

// ---------------- elementwise utility kernels ----------------

__global__ void k_zero(float* __restrict__ p, long n) {
  long i = (long)blockIdx.x * blockDim.x + threadIdx.x;
  if (i < n) p[i] = 0.0f;
}

__global__ void k_copy4(float4* __restrict__ dst, const float4* __restrict__ src, long n4) {
  long i = (long)blockIdx.x * blockDim.x + threadIdx.x;
  if (i < n4) dst[i] = src[i];
}

// ---------------- degree / normalization ----------------

__global__ void k_deg_bi(const int* __restrict__ eu, const int* __restrict__ ei,
                         float* __restrict__ deg, int E, int nUsers) {
  int e = blockIdx.x * blockDim.x + threadIdx.x;
  if (e < E) {
    atomicAdd(deg + eu[e], 1.0f);
    atomicAdd(deg + nUsers + ei[e], 1.0f);
  }
}

__global__ void k_deg_rows(const int* __restrict__ rows, float* __restrict__ deg, int E) {
  int e = blockIdx.x * blockDim.x + threadIdx.x;
  if (e < E) atomicAdd(deg + rows[e], 1.0f);
}

__global__ void k_finalize_dis(float* __restrict__ p, long n) {
  long i = (long)blockIdx.x * blockDim.x + threadIdx.x;
  if (i < n) {
    float d = p[i];
    p[i] = (d > 0.0f) ? rsqrtf(d) : 0.0f;   // d>=1 when >0, so max(d,1)==d
  }
}

// ---------------- SpMM scatter kernels (16 lanes per edge, float4 gathers) ----------------

__global__ void k_spmm_bi(const float* __restrict__ cur, float* __restrict__ nxt,
                          const float* __restrict__ dis,
                          const int* __restrict__ eu, const int* __restrict__ ei,
                          int E, int nUsers) {
  long t = (long)blockIdx.x * blockDim.x + threadIdx.x;
  int e = (int)(t >> 4);
  if (e >= E) return;
  int lane = (int)(t & 15);
  int u  = eu[e];
  int iv = nUsers + ei[e];
  float val = dis[u] * dis[iv];
  float4 xu = ((const float4*)(cur + (size_t)u  * FD))[lane];
  float4 xi = ((const float4*)(cur + (size_t)iv * FD))[lane];
  float* ou = nxt + (size_t)u  * FD + lane * 4;
  float* oi = nxt + (size_t)iv * FD + lane * 4;
  atomicAdd(ou + 0, val * xi.x);
  atomicAdd(ou + 1, val * xi.y);
  atomicAdd(ou + 2, val * xi.z);
  atomicAdd(ou + 3, val * xi.w);
  atomicAdd(oi + 0, val * xu.x);
  atomicAdd(oi + 1, val * xu.y);
  atomicAdd(oi + 2, val * xu.z);
  atomicAdd(oi + 3, val * xu.w);
}

__global__ void k_spmm_dir(const float* __restrict__ x, float* __restrict__ outp,
                           const float* __restrict__ dis,
                           const int* __restrict__ rows, const int* __restrict__ cols,
                           int E) {
  long t = (long)blockIdx.x * blockDim.x + threadIdx.x;
  int e = (int)(t >> 4);
  if (e >= E) return;
  int lane = (int)(t & 15);
  int r = rows[e];
  int c = cols[e];
  float val = dis[r] * dis[c];
  float4 xv = ((const float4*)(x + (size_t)c * FD))[lane];
  float* o = outp + (size_t)r * FD + lane * 4;
  atomicAdd(o + 0, val * xv.x);
  atomicAdd(o + 1, val * xv.y);
  atomicAdd(o + 2, val * xv.z);
  atomicAdd(o + 3, val * xv.w);
}

// ---------------- exact f32 tile add via V_WMMA_F32_16X16X4_F32 ----------------
// acc[16x16 tile] += add[16x16 tile], computed as sum_j I_j(16x4) x B_j(4x16) + C.
// Identity slices make every output element exactly one 1.0*x FMA -> bit-exact f32 add.
// One wave32 per tile; wave-uniform guard keeps EXEC all-ones as WMMA requires.
__global__ void k_wmma_add(float* __restrict__ acc, const float* __restrict__ add,
                           int nWaveTiles) {
  int wave = blockIdx.x * (blockDim.x >> 5) + ((int)threadIdx.x >> 5);
  if (wave >= nWaveTiles) return;
  int lane = (int)threadIdx.x & 31;
  int rowTile = wave >> 2;            // FD/16 == 4 column tiles per row block
  int colTile = wave & 3;
  int m    = lane & 15;
  int half = lane >> 4;               // lanes 16-31 hold the "upper K / upper M" halves
  size_t row0 = (size_t)rowTile * 16;
  int col = colTile * 16 + m;

  // C/D layout: VGPR r = row (r + half*8), column = lane&15 within the tile.
  float* cptr = acc + (row0 + (size_t)(half * 8)) * FD + col;
  v8f c;
#pragma unroll
  for (int r = 0; r < 8; ++r) c[r] = cptr[(size_t)r * FD];

  const float* bptr = add + row0 * FD + col;
#pragma unroll
  for (int j = 0; j < 4; ++j) {
    v2f a, b;
    int k0 = 4 * j + 2 * half;        // A: lanes<16 hold K=0,1 ; lanes>=16 hold K=2,3
    a[0] = (m == k0)     ? 1.0f : 0.0f;
    a[1] = (m == k0 + 1) ? 1.0f : 0.0f;
    b[0] = bptr[(size_t)k0 * FD];     // B: VGPR0 rows K=k0, VGPR1 rows K=k0+1
    b[1] = bptr[(size_t)(k0 + 1) * FD];
    c = __builtin_amdgcn_wmma_f32_16x16x4_f32(false, a, false, b, (short)0, c,
                                              false, false);
  }
#pragma unroll
  for (int r = 0; r < 8; ++r) cptr[(size_t)r * FD] = c[r];
}

__global__ void k_add_tail(float* __restrict__ acc, const float* __restrict__ add,
                           long start, long n) {
  long i = (long)blockIdx.x * blockDim.x + threadIdx.x;
  if (i < n) acc[start + i] += add[start + i];
}

// ---------------- host orchestration ----------------

static inline unsigned gridFor(long n, int b) { return (unsigned)((n + b - 1) / b); }

extern "C" void kernel_launch(void* const* d_in, const int* in_sizes, int n_in,
                              void* d_out, int out_size, void* d_ws, size_t ws_size,
                              hipStream_t stream) {
  const float* user_feat = (const float*)d_in[0];
  const float* item_feat = (const float*)d_in[1];
  const int* ui_u  = (const int*)d_in[2];
  const int* ui_i  = (const int*)d_in[3];
  const int* u_src = (const int*)d_in[4];
  const int* u_dst = (const int*)d_in[5];
  const int* i_src = (const int*)d_in[6];
  const int* i_dst = (const int*)d_in[7];

  const int NU = in_sizes[0] / FD;   // 100000
  const int NI = in_sizes[1] / FD;   // 50000
  const int NN = NU + NI;            // 150000
  const int E_ui = in_sizes[2];
  const int E_uu = in_sizes[4];
  const int E_ii = in_sizes[6];

  float* out = (float*)d_out;
  float* acc = out;                                  // ua ‖ ia  == full node matrix
  float* h2u = out + (size_t)NN * FD;
  float* h2i = h2u + (size_t)NU * FD;

  float* bufA  = (float*)d_ws;                       // NN*FD floats
  float* bufB  = bufA + (size_t)NN * FD;             // NN*FD floats
  float* dis_g = bufB + (size_t)NN * FD;             // NN
  float* dis_u = dis_g + NN;                         // NU
  float* dis_i = dis_u + NU;                         // NI  (dis arrays contiguous)

  const int B = 256;                                 // 8 wave32s per block
  const long nDis = (long)NN + NU + NI;

  // ---- symmetric normalization values (degree -> rsqrt) ----
  k_zero<<<gridFor(nDis, B), B, 0, stream>>>(dis_g, nDis);
  k_deg_bi<<<gridFor(E_ui, B), B, 0, stream>>>(ui_u, ui_i, dis_g, E_ui, NU);
  k_deg_rows<<<gridFor(E_uu, B), B, 0, stream>>>(u_src, dis_u, E_uu);
  k_deg_rows<<<gridFor(E_ii, B), B, 0, stream>>>(i_src, dis_i, E_ii);
  k_finalize_dis<<<gridFor(nDis, B), B, 0, stream>>>(dis_g, nDis);

  // ---- LightGCN: acc = emb; cur = emb ----
  const long nU4 = (long)NU * FD / 4, nI4 = (long)NI * FD / 4, nN4 = (long)NN * FD / 4;
  k_copy4<<<gridFor(nU4, B), B, 0, stream>>>((float4*)acc, (const float4*)user_feat, nU4);
  k_copy4<<<gridFor(nI4, B), B, 0, stream>>>((float4*)(acc + (size_t)NU * FD),
                                             (const float4*)item_feat, nI4);
  k_copy4<<<gridFor(nU4, B), B, 0, stream>>>((float4*)bufA, (const float4*)user_feat, nU4);
  k_copy4<<<gridFor(nI4, B), B, 0, stream>>>((float4*)(bufA + (size_t)NU * FD),
                                             (const float4*)item_feat, nI4);

  const int rowTiles = NN / 16;
  const int nWaveTiles = rowTiles * 4;               // 4 column tiles (FD=64)
  const long tailRows = (long)NN - (long)rowTiles * 16;

  float* cur = bufA;
  float* nxt = bufB;
  for (int layer = 0; layer < 2; ++layer) {
    // nxt = cur (residual), then scatter-add the propagation
    k_copy4<<<gridFor(nN4, B), B, 0, stream>>>((float4*)nxt, (const float4*)cur, nN4);
    k_spmm_bi<<<gridFor((long)E_ui * 16, B), B, 0, stream>>>(cur, nxt, dis_g,
                                                             ui_u, ui_i, E_ui, NU);
    // acc += nxt  (exact, on the matrix pipe)
    k_wmma_add<<<gridFor((long)nWaveTiles, B / 32), B, 0, stream>>>(acc, nxt, nWaveTiles);
    if (tailRows > 0) {
      long start = (long)rowTiles * 16 * FD, n = tailRows * FD;
      k_add_tail<<<gridFor(n, B), B, 0, stream>>>(acc, nxt, start, n);
    }
    float* t = cur; cur = nxt; nxt = t;
  }

  // ---- HAN user metapath: h2u = S_u * (S_u * user_feat) ----
  k_zero<<<gridFor((long)NU * FD, B), B, 0, stream>>>(bufA, (long)NU * FD);
  k_spmm_dir<<<gridFor((long)E_uu * 16, B), B, 0, stream>>>(user_feat, bufA, dis_u,
                                                            u_src, u_dst, E_uu);
  k_zero<<<gridFor((long)NU * FD, B), B, 0, stream>>>(h2u, (long)NU * FD);
  k_spmm_dir<<<gridFor((long)E_uu * 16, B), B, 0, stream>>>(bufA, h2u, dis_u,
                                                            u_src, u_dst, E_uu);

  // ---- HAN item metapath: h2i = S_i * (S_i * item_feat) ----
  k_zero<<<gridFor((long)NI * FD, B), B, 0, stream>>>(bufB, (long)NI * FD);
  k_spmm_dir<<<gridFor((long)E_ii * 16, B), B, 0, stream>>>(item_feat, bufB, dis_i,
                                                            i_src, i_dst, E_ii);
  k_zero<<<gridFor((long)NI * FD, B), B, 0, stream>>>(h2i, (long)NI * FD);
  k_spmm_dir<<<gridFor((long)E_ii * 16, B), B, 0, stream>>>(bufB, h2i, dis_i,
                                                            i_src, i_dst, E_ii);
}